// Block_53910429499877
// MI455X (gfx1250) — compile-verified
//
#include <hip/hip_runtime.h>
#include <hip/hip_bf16.h>
#include <math.h>

// ---------------------------------------------------------------------------
// Sizes (compile-time constants from the reference)
// ---------------------------------------------------------------------------
#define BATCH   64
#define HH      16
#define WW      16
#define FM      16
#define DIM     256          // H*W
#define NCH     33           // 1 + 2*FM
#define NTOK    (BATCH*NCH)  // 2112
#define HEADS   8
#define HD      32           // DIM/HEADS

typedef __attribute__((ext_vector_type(2))) float v2f;
typedef __attribute__((ext_vector_type(8))) float v8f;

// ---------------------------------------------------------------------------
// Kernel 1: morphology (erosion/dilation) + conv + BN + GELU for both branches
// one workgroup per batch, 256 threads (one per spatial position).
// Writes x2 channels 1..32 directly into d_out (they are final outputs too).
// ---------------------------------------------------------------------------
__global__ __launch_bounds__(256) void branch_kernel(
    const float* __restrict__ x, float* __restrict__ out,
    const float* __restrict__ se,  const float* __restrict__ sd,
    const float* __restrict__ s1w, const float* __restrict__ s1b,
    const float* __restrict__ s2w, const float* __restrict__ s2b,
    const float* __restrict__ sg,  const float* __restrict__ sbb,
    const float* __restrict__ sm,  const float* __restrict__ sv,
    const float* __restrict__ te,  const float* __restrict__ td,
    const float* __restrict__ t1w, const float* __restrict__ t1b,
    const float* __restrict__ t2w, const float* __restrict__ t2b,
    const float* __restrict__ tg,  const float* __restrict__ tbb,
    const float* __restrict__ tm,  const float* __restrict__ tv)
{
    __shared__ float rest[32 * DIM];   // 32 KB
    __shared__ float mE[FM * DIM];     // 16 KB  (erosion result)
    __shared__ float mD[FM * DIM];     // 16 KB  (dilation result)

    const int b   = blockIdx.x;
    const int tid = threadIdx.x;
    const float* xb = x + (size_t)b * NCH * DIM;

    for (int i = tid; i < 32 * DIM; i += 256) rest[i] = xb[DIM + i];
    __syncthreads();

    const int p = tid;
    const int h = p >> 4;
    const int w = p & 15;

    for (int br = 0; br < 2; ++br) {
        const float* eroW = br ? te : se;
        const float* dilW = br ? td : sd;

        float accE[FM], accD[FM];
#pragma unroll
        for (int oc = 0; oc < FM; ++oc) { accE[oc] = -1e30f; accD[oc] = -1e30f; }

        for (int c = 0; c < 32; ++c) {
            for (int kh = 0; kh < 3; ++kh) {
                const int hh = h + kh - 1;
                for (int kw = 0; kw < 3; ++kw) {
                    const int ww = w + kw - 1;
                    const float pv = (hh >= 0 && hh < HH && ww >= 0 && ww < WW)
                                   ? rest[c * DIM + hh * WW + ww] : 0.f;
                    const int widx = c * 9 + kh * 3 + kw;   // (C,3,3) inner index
#pragma unroll
                    for (int oc = 0; oc < FM; ++oc) {
                        const float we = eroW[oc * 288 + widx];   // uniform -> SGPR load
                        const float wd = dilW[oc * 288 + widx];
                        accE[oc] = fmaxf(accE[oc], we - pv);
                        accD[oc] = fmaxf(accD[oc], wd + pv);
                    }
                }
            }
        }

        __syncthreads();   // previous branch's conv reads are done
#pragma unroll
        for (int oc = 0; oc < FM; ++oc) {
            mE[oc * DIM + p] = -accE[oc];   // erosion = -max(w - p)
            mD[oc * DIM + p] =  accD[oc];   // dilation = max(w + p)
        }
        __syncthreads();

        const float* c1w = br ? t1w : s1w;
        const float* c1b = br ? t1b : s1b;
        const float* c2w = br ? t2w : s2w;
        const float* c2b = br ? t2b : s2b;
        const float* bg  = br ? tg  : sg;
        const float* bb  = br ? tbb : sbb;
        const float* bm  = br ? tm  : sm;
        const float* bv  = br ? tv  : sv;

        for (int oc = 0; oc < FM; ++oc) {
            float z = c1b[oc] + c2b[oc];
            if (br == 0) {           // 1x1 conv, pad 0
                for (int c = 0; c < FM; ++c) {
                    z += c1w[oc * FM + c] * mE[c * DIM + p];
                    z += c2w[oc * FM + c] * mD[c * DIM + p];
                }
            } else {                 // 3x3 conv, pad 1
                for (int c = 0; c < FM; ++c) {
                    for (int kh = 0; kh < 3; ++kh) {
                        const int hh = h + kh - 1;
                        if (hh < 0 || hh >= HH) continue;
                        for (int kw = 0; kw < 3; ++kw) {
                            const int ww = w + kw - 1;
                            if (ww < 0 || ww >= WW) continue;
                            const int wi = ((oc * FM + c) * 3 + kh) * 3 + kw;
                            const int si = c * DIM + hh * WW + ww;
                            z += c1w[wi] * mE[si];
                            z += c2w[wi] * mD[si];
                        }
                    }
                }
            }
            // BatchNorm (inference) + exact GELU
            const float inv = rsqrtf(bv[oc] + 1e-5f);
            float y = (z - bm[oc]) * bg[oc] * inv + bb[oc];
            y = 0.5f * y * (1.f + erff(y * 0.70710678118654752f));
            out[(size_t)b * NCH * DIM + (size_t)(1 + br * FM + oc) * DIM + p] = y;
        }
    }
}

// ---------------------------------------------------------------------------
// Kernel 2: LayerNorm of all 2112 tokens (one wave32 per token, 8 vals/lane)
// token source: channel 0 from x, channels 1..32 from d_out (written above).
// ---------------------------------------------------------------------------
__global__ __launch_bounds__(256) void ln_kernel(
    const float* __restrict__ x, const float* __restrict__ x2,
    const float* __restrict__ g, const float* __restrict__ bvec,
    float* __restrict__ xn)
{
    const int wave = threadIdx.x >> 5;
    const int lane = threadIdx.x & 31;
    const int j = blockIdx.x * 8 + wave;       // 0..2111 (grid exact)
    const int ch = j % NCH;
    const float* src = ((ch == 0) ? x : x2) + (size_t)j * DIM;

    float v[8], s = 0.f, s2 = 0.f;
#pragma unroll
    for (int i = 0; i < 8; ++i) {
        v[i] = src[lane + 32 * i];
        s += v[i]; s2 += v[i] * v[i];
    }
#pragma unroll
    for (int o = 16; o > 0; o >>= 1) {
        s  += __shfl_xor(s,  o, 32);
        s2 += __shfl_xor(s2, o, 32);
    }
    const float mu  = s * (1.f / DIM);
    const float var = s2 * (1.f / DIM) - mu * mu;
    const float inv = rsqrtf(var + 1e-6f);
#pragma unroll
    for (int i = 0; i < 8; ++i) {
        const int d = lane + 32 * i;
        xn[(size_t)j * DIM + d] = (v[i] - mu) * inv * g[d] + bvec[d];
    }
}

// ---------------------------------------------------------------------------
// Kernel 3: fp32 WMMA GEMM  Out[M,N] = A[M,K] * W^T  (out = xn @ W.T)
// M = 2112, N = 256, K = 256.  V_WMMA_F32_16X16X4_F32, 64 k-steps per tile.
// Block: 256 thr = 8 waves, each wave one 16x16 M-tile; 16 shared N-columns
// of W staged in LDS as bpanel[k][n].
// ---------------------------------------------------------------------------
__global__ __launch_bounds__(256) void gemm_wmma(
    const float* __restrict__ A, const float* __restrict__ W,
    float* __restrict__ Out)
{
    __shared__ float bpanel[256 * 16];   // [k][n], 16 KB
    const int tid   = threadIdx.x;
    const int nbase = blockIdx.x * 16;

    {   // stage B panel: B[k][n] = W[(nbase+n)*256 + k]
        const int n  = tid >> 4;
        const int k0 = tid & 15;
        const float* wr = W + (size_t)(nbase + n) * 256;
#pragma unroll
        for (int i = 0; i < 16; ++i) {
            const int k = k0 + 16 * i;
            bpanel[k * 16 + n] = wr[k];
        }
    }
    __syncthreads();

    const int wave  = tid >> 5;
    const int lane  = tid & 31;
    const int mtile = blockIdx.y * 8 + wave;
    if (mtile < 132) {                     // wave-uniform: EXEC stays all-ones
        const int l16  = lane & 15;
        const int half = lane >> 4;
        // A 16x4 layout: lanes 0-15 -> K={k0,k0+1}, lanes 16-31 -> K={k0+2,k0+3}
        const float* arow = A + (size_t)(mtile * 16 + l16) * 256 + 2 * half;
        v8f c = {};
        for (int k0 = 0; k0 < 256; k0 += 4) {
            v2f a;
            a[0] = arow[k0];
            a[1] = arow[k0 + 1];
            // B 4x16 layout: VGPR0 rows k0 / k0+1 on lane halves, VGPR1 rows k0+2 / k0+3
            v2f bv;
            bv[0] = bpanel[(k0 + half) * 16 + l16];
            bv[1] = bpanel[(k0 + 2 + half) * 16 + l16];
            c = __builtin_amdgcn_wmma_f32_16x16x4_f32(
                    false, a, false, bv, (short)0, c, false, false);
        }
        // D layout: VGPR r -> rows r (lanes 0-15) / r+8 (lanes 16-31), N = l16
#pragma unroll
        for (int r = 0; r < 8; ++r) {
            const int row = mtile * 16 + r + 8 * half;
            Out[(size_t)row * 256 + nbase + l16] = c[r];
        }
    }
}

// ---------------------------------------------------------------------------
// Kernel 4: per-batch cross-attention (q proj, scores, softmax, V mix,
// output proj, residual, final LN) -> d_out channel 0.
// ---------------------------------------------------------------------------
__global__ __launch_bounds__(256) void attn_kernel(
    const float* __restrict__ xn, const float* __restrict__ Kp,
    const float* __restrict__ Vp, const float* __restrict__ x,
    const float* __restrict__ wq, const float* __restrict__ projw,
    const float* __restrict__ projb, const float* __restrict__ cng,
    const float* __restrict__ cnb, float* __restrict__ out)
{
    __shared__ float q[DIM];
    __shared__ float sc[HEADS][NCH];
    __shared__ float o[DIM];
    __shared__ float red[16];

    const int b = blockIdx.x;
    const int tid = threadIdx.x;
    const float* x0 = xn + (size_t)b * NCH * DIM;   // token 0 (normalized)

    // q = xn[:,0] @ wq.T
    {
        float acc = 0.f;
        const float* wr = wq + (size_t)tid * DIM;
        for (int k = 0; k < DIM; ++k) acc += x0[k] * wr[k];
        q[tid] = acc;
    }
    __syncthreads();

    // scores[h][t] = <q_h, k_h(t)> / sqrt(32)
    for (int idx = tid; idx < HEADS * NCH; idx += 256) {
        const int hh = idx / NCH, t = idx % NCH;
        const float* kp = Kp + ((size_t)(b * NCH + t)) * DIM + hh * HD;
        const float* qh = q + hh * HD;
        float s = 0.f;
        for (int d = 0; d < HD; ++d) s += qh[d] * kp[d];
        sc[hh][t] = s * 0.17677669529663687f;   // 32^-0.5
    }
    __syncthreads();

    if (tid < HEADS) {                          // softmax over 33 keys
        float mx = -1e30f;
        for (int t = 0; t < NCH; ++t) mx = fmaxf(mx, sc[tid][t]);
        float sm = 0.f;
        for (int t = 0; t < NCH; ++t) { const float e = expf(sc[tid][t] - mx); sc[tid][t] = e; sm += e; }
        const float r = 1.f / sm;
        for (int t = 0; t < NCH; ++t) sc[tid][t] *= r;
    }
    __syncthreads();

    {   // o = attn @ V
        const int hh = tid >> 5;
        float od = 0.f;
        for (int t = 0; t < NCH; ++t)
            od += sc[hh][t] * Vp[((size_t)(b * NCH + t)) * DIM + tid];
        o[tid] = od;
    }
    __syncthreads();

    // projection + residual with x2[:,0] (= x[:,0])
    float pv = projb[tid];
    {
        const float* pr = projw + (size_t)tid * DIM;
        for (int d = 0; d < DIM; ++d) pv += o[d] * pr[d];
    }
    const float cls = pv + x[(size_t)b * NCH * DIM + tid];

    // final LayerNorm over 256 (block reduce)
    float s = cls, s2 = cls * cls;
#pragma unroll
    for (int off = 16; off > 0; off >>= 1) {
        s  += __shfl_xor(s,  off, 32);
        s2 += __shfl_xor(s2, off, 32);
    }
    const int lane = tid & 31, wave = tid >> 5;
    if (lane == 0) { red[wave] = s; red[8 + wave] = s2; }
    __syncthreads();
    if (tid == 0) {
        float ts = 0.f, ts2 = 0.f;
        for (int i = 0; i < 8; ++i) { ts += red[i]; ts2 += red[8 + i]; }
        red[0] = ts; red[8] = ts2;
    }
    __syncthreads();
    const float mu  = red[0] * (1.f / DIM);
    const float var = red[8] * (1.f / DIM) - mu * mu;
    const float inv = rsqrtf(var + 1e-6f);
    out[(size_t)b * NCH * DIM + tid] = (cls - mu) * inv * cng[tid] + cnb[tid];
}

// ---------------------------------------------------------------------------
extern "C" void kernel_launch(void* const* d_in, const int* in_sizes, int n_in,
                              void* d_out, int out_size, void* d_ws, size_t ws_size,
                              hipStream_t stream)
{
    const float* x      = (const float*)d_in[0];
    const float* an_g   = (const float*)d_in[1];
    const float* an_b   = (const float*)d_in[2];
    const float* cn_g   = (const float*)d_in[3];
    const float* cn_b   = (const float*)d_in[4];
    const float* wq     = (const float*)d_in[5];
    const float* wk     = (const float*)d_in[6];
    const float* wv     = (const float*)d_in[7];
    const float* projw  = (const float*)d_in[8];
    const float* projb  = (const float*)d_in[9];
    const float* se     = (const float*)d_in[10];
    const float* sd     = (const float*)d_in[11];
    const float* s1w    = (const float*)d_in[12];
    const float* s1b    = (const float*)d_in[13];
    const float* s2w    = (const float*)d_in[14];
    const float* s2b    = (const float*)d_in[15];
    const float* sg     = (const float*)d_in[16];
    const float* sbb    = (const float*)d_in[17];
    const float* sm     = (const float*)d_in[18];
    const float* sv     = (const float*)d_in[19];
    const float* te     = (const float*)d_in[20];
    const float* td     = (const float*)d_in[21];
    const float* t1w    = (const float*)d_in[22];
    const float* t1b    = (const float*)d_in[23];
    const float* t2w    = (const float*)d_in[24];
    const float* t2b    = (const float*)d_in[25];
    const float* tg     = (const float*)d_in[26];
    const float* tbb    = (const float*)d_in[27];
    const float* tm     = (const float*)d_in[28];
    const float* tv     = (const float*)d_in[29];

    float* out = (float*)d_out;
    float* xn  = (float*)d_ws;                 // 2112*256 f32
    float* Kp  = xn + (size_t)NTOK * DIM;      // 2112*256 f32
    float* Vp  = Kp + (size_t)NTOK * DIM;      // 2112*256 f32

    // 1. branches -> out channels 1..32 (also x2[1:])
    branch_kernel<<<BATCH, 256, 0, stream>>>(x, out,
        se, sd, s1w, s1b, s2w, s2b, sg, sbb, sm, sv,
        te, td, t1w, t1b, t2w, t2b, tg, tbb, tm, tv);

    // 2. LN of all tokens -> xn
    ln_kernel<<<NTOK / 8, 256, 0, stream>>>(x, out, an_g, an_b, xn);

    // 3. K / V projections via fp32 WMMA
    gemm_wmma<<<dim3(16, 17), 256, 0, stream>>>(xn, wk, Kp);
    gemm_wmma<<<dim3(16, 17), 256, 0, stream>>>(xn, wv, Vp);

    // 4. attention + final LN -> out channel 0
    attn_kernel<<<BATCH, 256, 0, stream>>>(xn, Kp, Vp, x, wq,
        projw, projb, cn_g, cn_b, out);
}